// PyTorchDWT2D_61770219651373
// MI455X (gfx1250) — compile-verified
//
#include <hip/hip_runtime.h>

// 2-D Haar DWT, one level, on (8,64,512,512) f32 -> 4 x (8,64,256,256) f32.
// Pure streaming problem (1 GiB traffic, ~0.4 GFLOP): optimize the memory
// path on MI455X — async global->LDS b128 staging (ASYNCcnt), conflict-free
// LDS reads, non-temporal coalesced stores. WMMA is deliberately NOT used:
// it cannot reduce bytes moved and would only add cross-lane transposes.

namespace {

constexpr int kH = 512;
constexpr int kW = 512;
constexpr int kWOut = 256;
constexpr int kRowsInPerBlock = 16;   // input rows staged per block (32 KB)
constexpr int kRowsOutPerBlock = 8;   // output rows per block
constexpr int kThreads = 256;
constexpr float kInvSqrt2 = 0.70710678118654752440f;  // np.float32(1/sqrt(2))

// GNU-style vector type to match the builtin's expected parameter type
// (diagnostic showed: "__attribute__((__vector_size__(4 * sizeof(int)))) int __device__ *").
typedef int v4i __attribute__((vector_size(16)));
typedef __attribute__((address_space(1))) v4i* global_v4i_ptr;
typedef __attribute__((address_space(3))) v4i* lds_v4i_ptr;

__device__ __forceinline__ void async_global_to_lds_b128(const float* g, float* l) {
#if __has_builtin(__builtin_amdgcn_global_load_async_to_lds_b128)
  __builtin_amdgcn_global_load_async_to_lds_b128(
      (global_v4i_ptr)(unsigned long long)g,
      (lds_v4i_ptr)(unsigned int)(unsigned long long)l,
      /*offset=*/0, /*cpol=*/0);
#else
  // VDST = LDS byte offset (low 32 bits of the flat shared address),
  // VADDR = 64-bit global address, no SADDR.
  unsigned lds_off = (unsigned)(unsigned long long)l;
  asm volatile("global_load_async_to_lds_b128 %0, %1, off"
               :
               : "v"(lds_off), "v"(g)
               : "memory");
#endif
}

__device__ __forceinline__ void wait_asynccnt_zero() {
#if __has_builtin(__builtin_amdgcn_s_wait_asynccnt)
  __builtin_amdgcn_s_wait_asynccnt(0);
#else
  asm volatile("s_wait_asynccnt 0" ::: "memory");
#endif
}

__global__ __launch_bounds__(kThreads) void haar_dwt2d_kernel(
    const float* __restrict__ x, float* __restrict__ out,
    unsigned long long band_stride) {
  __shared__ float smem[kRowsInPerBlock * kW];  // 32 KB

  const int img = blockIdx.x;  // fused (B,C) image index, 0..511
  const int rg  = blockIdx.y;  // row group, 0..31 (16 input rows each)
  const int t   = threadIdx.x; // 0..255 == output column

  const float* src = x + (unsigned long long)img * (kH * kW)
                       + (unsigned long long)rg * kRowsInPerBlock * kW;

  // Stage 16 contiguous input rows (one contiguous 32 KB span) into LDS.
  // 2048 16-byte chunks -> 8 async b128 instructions per thread, coalesced.
  constexpr int kChunksPerThread = (kRowsInPerBlock * kW) / 4 / kThreads;  // 8
#pragma unroll
  for (int k = 0; k < kChunksPerThread; ++k) {
    const int chunk = t + k * kThreads;
    async_global_to_lds_b128(src + chunk * 4, &smem[chunk * 4]);
  }
  wait_asynccnt_zero();
  __syncthreads();

  const unsigned long long obase =
      (unsigned long long)img * (kWOut * (kH / 2)) +
      (unsigned long long)(rg * kRowsOutPerBlock) * kWOut + (unsigned)t;

  float* out_ll = out;
  float* out_lh = out + band_stride;
  float* out_hl = out + 2 * band_stride;
  float* out_hh = out + 3 * band_stride;

#pragma unroll
  for (int k = 0; k < kRowsOutPerBlock; ++k) {
    // Lane t reads banks (2t)%64,(2t+1)%64: a wave covers all 64 banks once.
    const float2 r0 = *(const float2*)(&smem[(2 * k) * kW + 2 * t]);
    const float2 r1 = *(const float2*)(&smem[(2 * k + 1) * kW + 2 * t]);
    // Match the reference's rounding: two separate *INV_SQRT2 passes.
    const float xl_e = (r0.x + r0.y) * kInvSqrt2;  // W-lowpass, even H row
    const float xh_e = (r0.x - r0.y) * kInvSqrt2;  // W-highpass, even H row
    const float xl_o = (r1.x + r1.y) * kInvSqrt2;  // W-lowpass, odd H row
    const float xh_o = (r1.x - r1.y) * kInvSqrt2;  // W-highpass, odd H row
    const float ll = (xl_e + xl_o) * kInvSqrt2;
    const float lh = (xl_e - xl_o) * kInvSqrt2;
    const float hl = (xh_e + xh_o) * kInvSqrt2;
    const float hh = (xh_e - xh_o) * kInvSqrt2;
    const unsigned long long o = obase + (unsigned long long)k * kWOut;
    // Outputs are never re-read and exceed L2 -> non-temporal stores.
    __builtin_nontemporal_store(ll, out_ll + o);
    __builtin_nontemporal_store(lh, out_lh + o);
    __builtin_nontemporal_store(hl, out_hl + o);
    __builtin_nontemporal_store(hh, out_hh + o);
  }
}

}  // namespace

extern "C" void kernel_launch(void* const* d_in, const int* in_sizes, int n_in,
                              void* d_out, int out_size, void* d_ws, size_t ws_size,
                              hipStream_t stream) {
  (void)n_in; (void)out_size; (void)d_ws; (void)ws_size;
  const float* x = (const float*)d_in[0];
  float* out = (float*)d_out;

  const int n_img = in_sizes[0] / (kH * kW);  // 8*64 = 512
  const unsigned long long band_stride =
      (unsigned long long)n_img * (kH / 2) * (kW / 2);  // 33,554,432 elements

  dim3 grid((unsigned)n_img, (unsigned)(kH / kRowsInPerBlock));  // (512, 32)
  haar_dwt2d_kernel<<<grid, kThreads, 0, stream>>>(x, out, band_stride);
}